// ConvLSTMForecast_8083128451094
// MI455X (gfx1250) — compile-verified
//
#include <hip/hip_runtime.h>

typedef float v2f __attribute__((ext_vector_type(2)));
typedef float v8f __attribute__((ext_vector_type(8)));

#define B_    32
#define T_    64
#define X_    4096
#define HID_  16
#define K_    5
#define PAD_  2
#define FUT_  10
#define CIN_  17           // C + HID
#define KTOT  85           // CIN_ * K_      (real K)
#define KP_   8            // padded taps per input channel (5 real + 3 zero)
#define NCHP  34           // CIN_*KP_/4 = 136/4 K-chunks of 4 for f32 WMMA
#define ROW_  136          // LDS row length per input channel (128 + halo + tap pad)

__device__ __forceinline__ float sigmoid_f(float x) {
    return 1.0f / (1.0f + __expf(-x));
}
__device__ __forceinline__ float tanh_f(float x) {
    float a = __expf(-2.0f * fabsf(x));
    float t = (1.0f - a) / (1.0f + a);
    return copysignf(t, x);
}

__global__ void zero_f32(float* __restrict__ p, int n) {
    int i = blockIdx.x * blockDim.x + threadIdx.x;
    if (i < n) p[i] = 0.0f;
}

// One ConvLSTM cell step.
// Grid: 1024 blocks x 256 threads. Block = 8 wave32 tiles = 128 consecutive x
// of one batch row (b = blockIdx>>5, x0_blk = (blockIdx&31)*128).
// Each wave computes 16 positions x all 64 gate channels with f32 WMMA.
__global__ __launch_bounds__(256) void cell_step(
    const float* __restrict__ xsrc, int x_bstride,
    const float* __restrict__ W,    const float* __restrict__ bias,
    const float* __restrict__ h_in, const float* __restrict__ c_in,
    float* __restrict__ h_out,      float* __restrict__ c_out)
{
    __shared__ float ldsW[4 * NCHP * 64];   // A-fragments: [gate][chunk][lane][2]
    __shared__ float ldsI[CIN_ * ROW_];     // input slab:  [ci][j], j = p + tap

    int tid = threadIdx.x;
    int b      = blockIdx.x >> 5;
    int x0_blk = (blockIdx.x & 31) << 7;

    const float* xp = xsrc + b * x_bstride;
    const float* hp = h_in + b * (HID_ * X_);

    // --- Stage weights as WMMA A-fragments (K padded to 8 slots/channel) ---
    for (int e = tid; e < 4 * NCHP * 32; e += 256) {
        int lane  = e & 31;
        int chunk = (e >> 5) % NCHP;
        int G     = e / (32 * NCHP);
        int m  = lane & 15;                    // output channel within gate
        int kb = chunk * 4 + (lane >> 4) * 2;  // K pair for this lane half
        int ci   = kb >> 3;                    // kb, kb+1 share ci (kb&7 <= 6)
        int tap0 = kb & 7;
        float w0 = (tap0     < K_) ? W[(G * 16 + m) * KTOT + ci * K_ + tap0    ] : 0.0f;
        float w1 = (tap0 + 1 < K_) ? W[(G * 16 + m) * KTOT + ci * K_ + tap0 + 1] : 0.0f;
        ldsW[e * 2]     = w0;
        ldsW[e * 2 + 1] = w1;
    }
    // --- Stage input slab: ldsI[ci][j] = in[ci][x0_blk - PAD + j], zero-padded ---
    for (int e = tid; e < CIN_ * ROW_; e += 256) {
        int ci = e / ROW_;
        int j  = e - ci * ROW_;
        int gx = x0_blk - PAD_ + j;
        float v = 0.0f;
        if (gx >= 0 && gx < X_)
            v = (ci == 0) ? xp[gx] : hp[(ci - 1) * X_ + gx];
        ldsI[e] = v;
    }
    __syncthreads();

    int wave = tid >> 5;
    int lane = tid & 31;
    int n    = lane & 15;       // B-matrix column (position within 16-tile)
    int half = lane >> 4;

    // j base for this lane: p + 2*half  (p = position within block slab)
    int jbase = wave * 16 + n + 2 * half;

    v8f acc0 = {}; v8f acc1 = {}; v8f acc2 = {}; v8f acc3 = {};

    for (int ch = 0; ch < NCHP; ++ch) {
        int ci = ch >> 1;
        int j0 = ci * ROW_ + ((ch & 1) << 2) + jbase;
        v2f bf;
        bf[0] = ldsI[j0];       // two consecutive taps -> ds_load_2addr_b32
        bf[1] = ldsI[j0 + 1];
        const float* wl = &ldsW[(ch * 32 + lane) * 2];
        v2f a0 = *(const v2f*)(wl + 0 * NCHP * 64);
        v2f a1 = *(const v2f*)(wl + 1 * NCHP * 64);
        v2f a2 = *(const v2f*)(wl + 2 * NCHP * 64);
        v2f a3 = *(const v2f*)(wl + 3 * NCHP * 64);
        acc0 = __builtin_amdgcn_wmma_f32_16x16x4_f32(false, a0, false, bf, (short)0, acc0, false, false);
        acc1 = __builtin_amdgcn_wmma_f32_16x16x4_f32(false, a1, false, bf, (short)0, acc1, false, false);
        acc2 = __builtin_amdgcn_wmma_f32_16x16x4_f32(false, a2, false, bf, (short)0, acc2, false, false);
        acc3 = __builtin_amdgcn_wmma_f32_16x16x4_f32(false, a3, false, bf, (short)0, acc3, false, false);
    }

    // Gate math directly on D-fragment layout:
    // acc[G][r] at lane L = cc[gate G][chan r + 8*(L/16)][x0 + L%16]
    const float* cin = c_in + b * (HID_ * X_);
    float* ho = h_out + b * (HID_ * X_);
    float* co = c_out + b * (HID_ * X_);
    int x = x0_blk + wave * 16 + n;
    #pragma unroll
    for (int r = 0; r < 8; ++r) {
        int chn = r + 8 * half;
        float gi = acc0[r] + bias[ 0 + chn];
        float gf = acc1[r] + bias[16 + chn];
        float go = acc2[r] + bias[32 + chn];
        float gg = acc3[r] + bias[48 + chn];
        int idx = chn * X_ + x;
        float cn = sigmoid_f(gf) * cin[idx] + sigmoid_f(gi) * tanh_f(gg);
        float hn = sigmoid_f(go) * tanh_f(cn);
        co[idx] = cn;
        ho[idx] = hn;
    }
}

// Decoder head: out = tanh(1x1 conv over HID channels + bias); also writes feedback x.
__global__ __launch_bounds__(256) void proj_out(
    const float* __restrict__ h, const float* __restrict__ Wout,
    const float* __restrict__ bout, float* __restrict__ out,
    float* __restrict__ fb, int f)
{
    int i = blockIdx.x * 256 + threadIdx.x;  // 0 .. B_*X_-1
    if (i >= B_ * X_) return;
    int b = i >> 12;
    int x = i & (X_ - 1);
    const float* hp = h + b * (HID_ * X_) + x;
    float s = bout[0];
    #pragma unroll
    for (int chn = 0; chn < HID_; ++chn) s += Wout[chn] * hp[chn * X_];
    float o = tanh_f(s);
    out[(b * FUT_ + f) * X_ + x] = o;        // preds[b, f, 0, x, 0]
    fb[i] = o;
}

extern "C" void kernel_launch(void* const* d_in, const int* in_sizes, int n_in,
                              void* d_out, int out_size, void* d_ws, size_t ws_size,
                              hipStream_t stream) {
    (void)in_sizes; (void)n_in; (void)out_size; (void)ws_size;
    const float* input_seq = (const float*)d_in[0];   // (B,T,1,X,1)
    const float* W_cell    = (const float*)d_in[1];   // (64,17,5,1)
    const float* b_cell    = (const float*)d_in[2];   // (64,)
    const float* W_out     = (const float*)d_in[3];   // (1,16,1,1)
    const float* b_out     = (const float*)d_in[4];   // (1,)
    // d_in[5] = future_steps (known constant FUT_=10)

    const int state_elems = B_ * HID_ * X_;           // 2,097,152 floats (8 MB)
    float* hA = (float*)d_ws;
    float* hB = hA + state_elems;
    float* cS = hB + state_elems;                     // cell state: pointwise, in-place safe
    float* fb = cS + state_elems;                     // decoder feedback (B*X)

    zero_f32<<<(state_elems + 255) / 256, 256, 0, stream>>>(hA, state_elems);
    zero_f32<<<(state_elems + 255) / 256, 256, 0, stream>>>(cS, state_elems);

    const float* hin = hA;
    float* hout = hB;
    // Encoder: 64 sequential steps, h double-buffered (cross-tile halo reads)
    for (int t = 0; t < T_; ++t) {
        cell_step<<<1024, 256, 0, stream>>>(input_seq + t * X_, T_ * X_,
                                            W_cell, b_cell, hin, cS, hout, cS);
        const float* tmp = hout; hout = (float*)hin; hin = tmp;
    }
    // Decoder: 10 steps, feedback through fb
    for (int f = 0; f < FUT_; ++f) {
        const float* xs = (f == 0) ? (input_seq + (T_ - 1) * X_) : fb;
        int xstride     = (f == 0) ? (T_ * X_) : X_;
        cell_step<<<1024, 256, 0, stream>>>(xs, xstride, W_cell, b_cell,
                                            hin, cS, hout, cS);
        const float* tmp = hout; hout = (float*)hin; hin = tmp;
        proj_out<<<(B_ * X_ + 255) / 256, 256, 0, stream>>>(hin, W_out, b_out,
                                                            (float*)d_out, fb, f);
    }
}